// GraphEncoder_5394478924647
// MI455X (gfx1250) — compile-verified
//
#include <hip/hip_runtime.h>

#define N_NODES_C 262144
#define N_EDGES_C 4194304
#define N_GRAPHS_C 4096
#define F_IN_C 32
#define D_C 16
#define H_C 32
#define OUT_C 27
#define EPS_C 1e-12f

typedef __attribute__((ext_vector_type(2))) float v2f;
typedef __attribute__((ext_vector_type(8))) float v8f;

// One V_WMMA_F32_16X16X4_F32 step: C += A(16xK slice)[kb..kb+3] * B(Kx16 slice).
// A layout: lane l, reg r -> row = l&15, K = kb + r + 2*(l>>4)
// B layout: lane l, reg r -> col = ncol + (l&15), K = kb + r + 2*(l>>4)
__device__ __forceinline__ v8f wmma_f32_step(const float* __restrict__ Abase, int lda, int kb,
                                             const float* __restrict__ Bbase, int ldb, int ncol,
                                             int lid, int half, v8f c) {
    v2f a, b;
    const float* arow = Abase + lid * lda + kb + 2 * half;
    a.x = arow[0];
    a.y = arow[1];
    int k0 = kb + 2 * half;
    b.x = Bbase[(k0 + 0) * ldb + ncol + lid];
    b.y = Bbase[(k0 + 1) * ldb + ncol + lid];
    return __builtin_amdgcn_wmma_f32_16x16x4_f32(false, a, false, b, (short)0, c, false, false);
}

// h = relu(X @ W_node + b_node) : [N,32] @ [32,16] -> [N,16]
__global__ void k_node_init(const float* __restrict__ X, const float* __restrict__ W,
                            const float* __restrict__ bias, float* __restrict__ h) {
    int wave = blockIdx.x * (blockDim.x >> 5) + (threadIdx.x >> 5);
    int lane = threadIdx.x & 31;
    int lid = lane & 15, half = lane >> 4;
    int base = wave * 16;
    const float* A = X + (size_t)base * F_IN_C;
    v8f c = {0.f, 0.f, 0.f, 0.f, 0.f, 0.f, 0.f, 0.f};
#pragma unroll
    for (int kb = 0; kb < F_IN_C; kb += 4)
        c = wmma_f32_step(A, F_IN_C, kb, W, D_C, 0, lid, half, c);
    float bv = bias[lid];
#pragma unroll
    for (int r = 0; r < 8; ++r) {
        float v = c[r] + bv;
        h[(size_t)(base + r + 8 * half) * D_C + lid] = v > 0.f ? v : 0.f;
    }
}

// m = relu(h @ W_hidden) : [N,16] @ [16,32] -> [N,32]  (two 16-col WMMA blocks)
__global__ void k_msg(const float* __restrict__ h, const float* __restrict__ Wh,
                      float* __restrict__ m) {
    int wave = blockIdx.x * (blockDim.x >> 5) + (threadIdx.x >> 5);
    int lane = threadIdx.x & 31;
    int lid = lane & 15, half = lane >> 4;
    int base = wave * 16;
    const float* A = h + (size_t)base * D_C;
    v8f c0 = {0.f, 0.f, 0.f, 0.f, 0.f, 0.f, 0.f, 0.f};
    v8f c1 = {0.f, 0.f, 0.f, 0.f, 0.f, 0.f, 0.f, 0.f};
#pragma unroll
    for (int kb = 0; kb < D_C; kb += 4) {
        c0 = wmma_f32_step(A, D_C, kb, Wh, H_C, 0, lid, half, c0);
        c1 = wmma_f32_step(A, D_C, kb, Wh, H_C, 16, lid, half, c1);
    }
#pragma unroll
    for (int r = 0; r < 8; ++r) {
        size_t row = (size_t)(base + r + 8 * half) * H_C;
        float v0 = c0[r], v1 = c1[r];
        m[row + lid] = v0 > 0.f ? v0 : 0.f;
        m[row + 16 + lid] = v1 > 0.f ? v1 : 0.f;
    }
}

__global__ void k_zero(float* __restrict__ p, int n) {
    int i = blockIdx.x * blockDim.x + threadIdx.x;
    if (i < n) p[i] = 0.f;
}

// pooled[tgt] += m[src] : 8 lanes per edge, float4 each (L2-resident f32 atomics)
__global__ void k_scatter(const float* __restrict__ m, const int* __restrict__ src,
                          const int* __restrict__ tgt, float* __restrict__ pooled) {
    int t = blockIdx.x * blockDim.x + threadIdx.x;
    int e = t >> 3;
    int c4 = (t & 7) << 2;
    int s = src[e];
    int d = tgt[e];
    const float4 v = *reinterpret_cast<const float4*>(m + (size_t)s * H_C + c4);
    float* p = pooled + (size_t)d * H_C + c4;
    unsafeAtomicAdd(p + 0, v.x);
    unsafeAtomicAdd(p + 1, v.y);
    unsafeAtomicAdd(p + 2, v.z);
    unsafeAtomicAdd(p + 3, v.w);
}

// h = l2norm(relu(pooled @ W_pool + h @ W_self)) in place, per 16-node tile
__global__ void k_combine(const float* __restrict__ pooled, const float* __restrict__ Wp,
                          const float* __restrict__ Ws, float* __restrict__ h) {
    int wave = blockIdx.x * (blockDim.x >> 5) + (threadIdx.x >> 5);
    int lane = threadIdx.x & 31;
    int lid = lane & 15, half = lane >> 4;
    int base = wave * 16;
    v8f c = {0.f, 0.f, 0.f, 0.f, 0.f, 0.f, 0.f, 0.f};
    const float* Ap = pooled + (size_t)base * H_C;
#pragma unroll
    for (int kb = 0; kb < H_C; kb += 4)
        c = wmma_f32_step(Ap, H_C, kb, Wp, D_C, 0, lid, half, c);
    const float* Ah = h + (size_t)base * D_C;
#pragma unroll
    for (int kb = 0; kb < D_C; kb += 4)
        c = wmma_f32_step(Ah, D_C, kb, Ws, D_C, 0, lid, half, c);
#pragma unroll
    for (int r = 0; r < 8; ++r) {
        float v = c[r];
        c[r] = v > 0.f ? v : 0.f;
    }
    // l2-normalize each node's 16 features (spread across a 16-lane half-group)
#pragma unroll
    for (int r = 0; r < 8; ++r) {
        float ss = c[r] * c[r];
        ss += __shfl_xor(ss, 1, 16);
        ss += __shfl_xor(ss, 2, 16);
        ss += __shfl_xor(ss, 4, 16);
        ss += __shfl_xor(ss, 8, 16);
        float sc = rsqrtf(fmaxf(ss, EPS_C));
        h[(size_t)(base + r + 8 * half) * D_C + lid] = c[r] * sc;
    }
}

// readout[g] += h[node] (per-graph sum pool); one thread per (node, feature)
__global__ void k_readout(const float* __restrict__ h, const int* __restrict__ gid,
                          float* __restrict__ ro) {
    int i = blockIdx.x * blockDim.x + threadIdx.x;
    int node = i >> 4;
    int f = i & 15;
    int g = gid[node];
    unsafeAtomicAdd(ro + (size_t)g * D_C + f, h[i]);
}

// out = readout @ W_out + b_out : [G,16] @ [16,27]
__global__ void k_out(const float* __restrict__ ro, const float* __restrict__ Wo,
                      const float* __restrict__ bo, float* __restrict__ out) {
    int i = blockIdx.x * blockDim.x + threadIdx.x;
    if (i >= N_GRAPHS_C * OUT_C) return;
    int g = i / OUT_C;
    int o = i - g * OUT_C;
    float acc = bo[o];
#pragma unroll
    for (int f = 0; f < D_C; ++f)
        acc += ro[(size_t)g * D_C + f] * Wo[f * OUT_C + o];
    out[i] = acc;
}

extern "C" void kernel_launch(void* const* d_in, const int* in_sizes, int n_in,
                              void* d_out, int out_size, void* d_ws, size_t ws_size,
                              hipStream_t stream) {
    (void)in_sizes; (void)n_in; (void)out_size; (void)ws_size;
    const float* X        = (const float*)d_in[0];
    const float* W_node   = (const float*)d_in[1];
    const float* b_node   = (const float*)d_in[2];
    const float* W_hidden = (const float*)d_in[3];
    const float* W_pool   = (const float*)d_in[4];
    const float* W_self   = (const float*)d_in[5];
    const float* W_out    = (const float*)d_in[6];
    const float* b_out    = (const float*)d_in[7];
    const int* esrc       = (const int*)d_in[8];
    const int* etgt       = (const int*)d_in[9];
    const int* gid        = (const int*)d_in[10];

    float* ws = (float*)d_ws;
    float* h      = ws;                                   // N*16  (16 MB)
    float* m      = h + (size_t)N_NODES_C * D_C;          // N*32  (32 MB)
    float* pooled = m + (size_t)N_NODES_C * H_C;          // N*32  (32 MB)
    float* ro     = pooled + (size_t)N_NODES_C * H_C;     // G*16  (256 KB)

    dim3 blk(256);
    const int tileBlocks = N_NODES_C / 16 / 8;            // 8 waves/block, 16 nodes/wave

    k_node_init<<<tileBlocks, blk, 0, stream>>>(X, W_node, b_node, h);
    for (int round = 0; round < 2; ++round) {
        k_msg<<<tileBlocks, blk, 0, stream>>>(h, W_hidden, m);
        k_zero<<<(N_NODES_C * H_C) / 256, blk, 0, stream>>>(pooled, N_NODES_C * H_C);
        k_scatter<<<(N_EDGES_C * 8) / 256, blk, 0, stream>>>(m, esrc, etgt, pooled);
        k_combine<<<tileBlocks, blk, 0, stream>>>(pooled, W_pool, W_self, h);
    }
    k_zero<<<(N_GRAPHS_C * D_C) / 256, blk, 0, stream>>>(ro, N_GRAPHS_C * D_C);
    k_readout<<<(N_NODES_C * D_C) / 256, blk, 0, stream>>>(h, gid, ro);
    k_out<<<(N_GRAPHS_C * OUT_C + 255) / 256, blk, 0, stream>>>(ro, W_out, b_out, (float*)d_out);
}